// GATv2_15161234555390
// MI455X (gfx1250) — compile-verified
//
#include <hip/hip_runtime.h>
#include <hip/hip_bf16.h>
#include <math.h>

// ---------------------------------------------------------------------------
// GATv2 (2 layers) for MI455X / gfx1250, wave32, WMMA f32 16x16x4 for GEMMs.
// ---------------------------------------------------------------------------

typedef float v2f __attribute__((ext_vector_type(2)));
typedef float v8f __attribute__((ext_vector_type(8)));

#define F_IN   128
#define HEADS  4
#define CH     32
#define HC     128   // HEADS*CH
#define CLS    40
#define NEG_SLOPE 0.2f

// ---- monotone float <-> uint mapping for atomicMax on floats --------------
__device__ __forceinline__ unsigned fmu(float f) {
    unsigned u = __float_as_uint(f);
    return (u & 0x80000000u) ? ~u : (u | 0x80000000u);
}
__device__ __forceinline__ float mu2f(unsigned u) {
    return __uint_as_float((u & 0x80000000u) ? (u & 0x7FFFFFFFu) : ~u);
}
#define MU_NEG_INF 0x007FFFFFu   // fmu(-inf)

__device__ __forceinline__ float lrelu(float v) {
    return v > 0.0f ? v : NEG_SLOPE * v;
}

// ---------------------------------------------------------------------------
// fill kernel (workspace init each call; harness poisons buffers)
// ---------------------------------------------------------------------------
__global__ void fill_u32(unsigned* __restrict__ p, unsigned v, int n) {
    int i = blockIdx.x * blockDim.x + threadIdx.x;
    if (i < n) p[i] = v;
}

// ---------------------------------------------------------------------------
// WMMA f32 GEMM:  Out[nrows x NCOL] = X[nrows x K] @ W[K x NCOL]
// K, NCOL are compile-time: the K-loop fully unrolls into 32 back-to-back
// v_wmma_f32_16x16x4_f32 with immediate-offset global_load_b64 operands; no
// runtime address muls, no EXEC juggling in the hot path.
// One wave -> one 16x16 tile. grid.x = nrows/16, blockDim = 256 (8 waves =
// up to 8 column tiles; excess waves exit whole-wave, EXEC stays all-ones).
// ---------------------------------------------------------------------------
template<int K, int NCOL>
__global__ __launch_bounds__(256) void gemm16_wmma_f32(
    const float* __restrict__ X, const float* __restrict__ W,
    float* __restrict__ Out)
{
    constexpr int colTiles = (NCOL + 15) / 16;
    const int lane = threadIdx.x & 31;
    const int wave = threadIdx.x >> 5;
    if (wave >= colTiles) return;

    const int row0  = blockIdx.x << 4;
    const int lh    = lane & 15;          // M (for A) / N (for B)
    const int khalf = (lane >> 4) << 1;   // 0 for lanes 0-15, 2 for 16-31

    const float* xrow = X + (size_t)(row0 + lh) * K + khalf;
    const int c = (wave << 4) + lh;

    v8f acc = {};
    if constexpr (NCOL % 16 == 0) {
        // no column masking needed at all
        const float* wcol = W + (size_t)khalf * NCOL + c;
#pragma unroll
        for (int k = 0; k < K; k += 4) {
            v2f a, b;
            a.x = xrow[k];
            a.y = xrow[k + 1];
            b.x = wcol[(size_t)k * NCOL];
            b.y = wcol[(size_t)(k + 1) * NCOL];
            acc = __builtin_amdgcn_wmma_f32_16x16x4_f32(
                      false, a, false, b, (short)0, acc, false, false);
        }
        const int rbase = row0 + ((lane >> 4) << 3);
#pragma unroll
        for (int r = 0; r < 8; ++r)
            Out[(size_t)(rbase + r) * NCOL + c] = acc[r];
    } else {
        // ragged tail tile: clamp address (always valid), branch-free zeroing
        const bool cok = (c < NCOL);
        const int  cc  = cok ? c : (NCOL - 1);
        const float* wcol = W + (size_t)khalf * NCOL + cc;
#pragma unroll
        for (int k = 0; k < K; k += 4) {
            v2f a, b;
            a.x = xrow[k];
            a.y = xrow[k + 1];
            float b0 = wcol[(size_t)k * NCOL];
            float b1 = wcol[(size_t)(k + 1) * NCOL];
            b.x = cok ? b0 : 0.0f;
            b.y = cok ? b1 : 0.0f;
            acc = __builtin_amdgcn_wmma_f32_16x16x4_f32(
                      false, a, false, b, (short)0, acc, false, false);
        }
        if (cok) {
            const int rbase = row0 + ((lane >> 4) << 3);
#pragma unroll
            for (int r = 0; r < 8; ++r)
                Out[(size_t)(rbase + r) * NCOL + c] = acc[r];
        }
    }
}

// ---------------------------------------------------------------------------
// Layer 1 edge logits: wave per edge; lane = (head<<3)|quad, float4 per lane
// covers the 128 channels exactly. Reduce within groups of 8 lanes (one head).
// ---------------------------------------------------------------------------
__global__ __launch_bounds__(256) void edge_logits_h4(
    const int* __restrict__ ei, int E, int ET,
    const float* __restrict__ xl, const float* __restrict__ xr,
    const float* __restrict__ a1,                 // [4][32] = 128 floats
    float* __restrict__ eout,                     // [ET][4]
    unsigned* __restrict__ segmax)                // [N][4] monotone uints
{
    const int wid = (blockIdx.x * blockDim.x + threadIdx.x) >> 5;
    if (wid >= ET) return;
    const int lane = threadIdx.x & 31;

    int src, dst;
    if (wid < E) { src = ei[wid]; dst = ei[E + wid]; }
    else         { src = dst = wid - E; }

    const float4* xl4 = (const float4*)xl;
    const float4* xr4 = (const float4*)xr;
    const float4* a4  = (const float4*)a1;
    float4 l  = xl4[(size_t)src * 32 + lane];
    float4 r  = xr4[(size_t)dst * 32 + lane];
    float4 av = a4[lane];

    float s = av.x * lrelu(l.x + r.x) + av.y * lrelu(l.y + r.y)
            + av.z * lrelu(l.z + r.z) + av.w * lrelu(l.w + r.w);
    s += __shfl_xor(s, 1, 32);
    s += __shfl_xor(s, 2, 32);
    s += __shfl_xor(s, 4, 32);

    if ((lane & 7) == 0) {
        const int h = lane >> 3;
        eout[(size_t)wid * 4 + h] = s;
        atomicMax(&segmax[(size_t)dst * 4 + h], fmu(s));
    }
}

// ---------------------------------------------------------------------------
// Layer 1 exp + segment-sum: thread per (edge, head)
// ---------------------------------------------------------------------------
__global__ void expsum_h4(const int* __restrict__ ei, int E, int ET,
                          float* __restrict__ ebuf,
                          const unsigned* __restrict__ segmax,
                          float* __restrict__ segsum)
{
    int idx = blockIdx.x * blockDim.x + threadIdx.x;
    if (idx >= ET * 4) return;
    const int e = idx >> 2, h = idx & 3;
    const int dst = (e < E) ? ei[E + e] : (e - E);
    const float m = mu2f(segmax[(size_t)dst * 4 + h]);
    const float p = __expf(ebuf[idx] - m);
    ebuf[idx] = p;
    atomicAdd(&segsum[(size_t)dst * 4 + h], p);
}

// ---------------------------------------------------------------------------
// Layer 1 aggregation: wave per edge; 4 channels per lane, atomic scatter-add
// ---------------------------------------------------------------------------
__global__ __launch_bounds__(256) void aggregate_h4(
    const int* __restrict__ ei, int E, int ET,
    const float* __restrict__ ebuf, const float* __restrict__ segsum,
    const float* __restrict__ xl, float* __restrict__ accum)
{
    const int wid = (blockIdx.x * blockDim.x + threadIdx.x) >> 5;
    if (wid >= ET) return;
    const int lane = threadIdx.x & 31;

    int src, dst;
    if (wid < E) { src = ei[wid]; dst = ei[E + wid]; }
    else         { src = dst = wid - E; }

    const int h = lane >> 3;
    const float alpha = ebuf[(size_t)wid * 4 + h]
                      / segsum[(size_t)dst * 4 + h];

    const float4* xl4 = (const float4*)xl;
    float4 l = xl4[(size_t)src * 32 + lane];
    float* dp = accum + (size_t)dst * HC + lane * 4;
    atomicAdd(dp + 0, alpha * l.x);
    atomicAdd(dp + 1, alpha * l.y);
    atomicAdd(dp + 2, alpha * l.z);
    atomicAdd(dp + 3, alpha * l.w);
}

// ---------------------------------------------------------------------------
// ELU + bias (in place): h = elu(accum + b1)
// ---------------------------------------------------------------------------
__global__ void elu_bias(float* __restrict__ a, const float* __restrict__ b, int n)
{
    int idx = blockIdx.x * blockDim.x + threadIdx.x;
    if (idx >= n) return;
    float v = a[idx] + b[idx & (HC - 1)];
    a[idx] = v > 0.0f ? v : (__expf(v) - 1.0f);
}

// ---------------------------------------------------------------------------
// Layer 2 edge logits: wave per edge, H=1, C=40 (lane + lane+32 strip)
// ---------------------------------------------------------------------------
__global__ __launch_bounds__(256) void edge_logits_h1(
    const int* __restrict__ ei, int E, int ET,
    const float* __restrict__ xl, const float* __restrict__ xr,
    const float* __restrict__ a2,                 // 40 floats
    float* __restrict__ eout, unsigned* __restrict__ segmax)
{
    const int wid = (blockIdx.x * blockDim.x + threadIdx.x) >> 5;
    if (wid >= ET) return;
    const int lane = threadIdx.x & 31;

    int src, dst;
    if (wid < E) { src = ei[wid]; dst = ei[E + wid]; }
    else         { src = dst = wid - E; }

    float s;
    {
        float v = xl[(size_t)src * CLS + lane] + xr[(size_t)dst * CLS + lane];
        s = a2[lane] * lrelu(v);
    }
    if (lane < 8) {
        const int c2 = lane + 32;
        float v = xl[(size_t)src * CLS + c2] + xr[(size_t)dst * CLS + c2];
        s += a2[c2] * lrelu(v);
    }
    s += __shfl_xor(s, 1, 32);
    s += __shfl_xor(s, 2, 32);
    s += __shfl_xor(s, 4, 32);
    s += __shfl_xor(s, 8, 32);
    s += __shfl_xor(s, 16, 32);

    if (lane == 0) {
        eout[wid] = s;
        atomicMax(&segmax[dst], fmu(s));
    }
}

// ---------------------------------------------------------------------------
// Layer 2 exp + segment-sum: thread per edge
// ---------------------------------------------------------------------------
__global__ void expsum_h1(const int* __restrict__ ei, int E, int ET,
                          float* __restrict__ ebuf,
                          const unsigned* __restrict__ segmax,
                          float* __restrict__ segsum)
{
    int e = blockIdx.x * blockDim.x + threadIdx.x;
    if (e >= ET) return;
    const int dst = (e < E) ? ei[E + e] : (e - E);
    const float m = mu2f(segmax[dst]);
    const float p = __expf(ebuf[e] - m);
    ebuf[e] = p;
    atomicAdd(&segsum[dst], p);
}

// ---------------------------------------------------------------------------
// Layer 2 aggregation: wave per edge, 40 channels
// ---------------------------------------------------------------------------
__global__ __launch_bounds__(256) void aggregate_h1(
    const int* __restrict__ ei, int E, int ET,
    const float* __restrict__ ebuf, const float* __restrict__ segsum,
    const float* __restrict__ xl, float* __restrict__ accum)
{
    const int wid = (blockIdx.x * blockDim.x + threadIdx.x) >> 5;
    if (wid >= ET) return;
    const int lane = threadIdx.x & 31;

    int src, dst;
    if (wid < E) { src = ei[wid]; dst = ei[E + wid]; }
    else         { src = dst = wid - E; }

    const float alpha = ebuf[wid] / segsum[dst];
    atomicAdd(&accum[(size_t)dst * CLS + lane],
              alpha * xl[(size_t)src * CLS + lane]);
    if (lane < 8) {
        const int c2 = lane + 32;
        atomicAdd(&accum[(size_t)dst * CLS + c2],
                  alpha * xl[(size_t)src * CLS + c2]);
    }
}

// ---------------------------------------------------------------------------
// Final: out = log_softmax(accum2 + b2), wave per node over 40 classes
// ---------------------------------------------------------------------------
__global__ __launch_bounds__(256) void finalize_logsoftmax(
    const float* __restrict__ accum, const float* __restrict__ b2,
    float* __restrict__ out, int N)
{
    const int node = (blockIdx.x * blockDim.x + threadIdx.x) >> 5;
    if (node >= N) return;
    const int lane = threadIdx.x & 31;

    const float v1 = accum[(size_t)node * CLS + lane] + b2[lane];
    const float v2 = (lane < 8)
        ? accum[(size_t)node * CLS + 32 + lane] + b2[32 + lane]
        : -__int_as_float(0x7F800000);  // -inf for inactive strip

    float m = fmaxf(v1, v2);
    m = fmaxf(m, __shfl_xor(m, 1, 32));
    m = fmaxf(m, __shfl_xor(m, 2, 32));
    m = fmaxf(m, __shfl_xor(m, 4, 32));
    m = fmaxf(m, __shfl_xor(m, 8, 32));
    m = fmaxf(m, __shfl_xor(m, 16, 32));

    float se = __expf(v1 - m) + ((lane < 8) ? __expf(v2 - m) : 0.0f);
    se += __shfl_xor(se, 1, 32);
    se += __shfl_xor(se, 2, 32);
    se += __shfl_xor(se, 4, 32);
    se += __shfl_xor(se, 8, 32);
    se += __shfl_xor(se, 16, 32);

    const float ls = m + __logf(se);
    out[(size_t)node * CLS + lane] = v1 - ls;
    if (lane < 8)
        out[(size_t)node * CLS + 32 + lane] = v2 - ls;
}

// ---------------------------------------------------------------------------
// Host launcher
// ---------------------------------------------------------------------------
extern "C" void kernel_launch(void* const* d_in, const int* in_sizes, int n_in,
                              void* d_out, int out_size, void* d_ws, size_t ws_size,
                              hipStream_t stream)
{
    const float* x    = (const float*)d_in[0];
    const int*   ei   = (const int*)  d_in[1];   // [2][E] int32 (JAX x64 off)
    const float* w1l  = (const float*)d_in[2];
    const float* w1r  = (const float*)d_in[3];
    const float* a1   = (const float*)d_in[4];
    const float* b1   = (const float*)d_in[5];
    const float* w2l  = (const float*)d_in[6];
    const float* w2r  = (const float*)d_in[7];
    const float* a2   = (const float*)d_in[8];
    const float* b2   = (const float*)d_in[9];
    float* out = (float*)d_out;

    const int N  = in_sizes[0] / F_IN;   // 50000
    const int E  = in_sizes[1] / 2;      // 800000
    const int ET = E + N;                // edges + self loops

    // ---- workspace layout (bytes) ----
    char* ws = (char*)d_ws;
    size_t o = 0;
    float*    xl1  = (float*)(ws + o); o += (size_t)N * HC * 4;       // 25.6 MB
    float*    xr1  = (float*)(ws + o); o += (size_t)N * HC * 4;       // 25.6 MB
    float*    acc1 = (float*)(ws + o); o += (size_t)N * HC * 4;       // 25.6 MB (→ h after ELU)
    float*    e1   = (float*)(ws + o); o += (size_t)ET * 4 * 4;       // 13.6 MB
    unsigned* sm1  = (unsigned*)(ws + o); o += (size_t)N * 4 * 4;     //  0.8 MB
    float*    ss1  = (float*)(ws + o); o += (size_t)N * 4 * 4;        //  0.8 MB
    float*    acc2 = (float*)(ws + o); o += (size_t)N * CLS * 4;      //  8.0 MB
    float*    e2   = (float*)(ws + o); o += (size_t)ET * 4;           //  3.4 MB
    // layer-2 reuse of layer-1 buffers once they are dead:
    float*    xl2  = xl1;               // free after aggregate_h4
    float*    xr2  = xr1;
    unsigned* sm2  = sm1;               // free after expsum_h4 / aggregate_h4
    float*    ss2  = ss1;
    (void)ws_size; (void)n_in; (void)out_size;

    const int B = 256;
    const int rowTiles   = N / 16;                  // 3125, N is 16-aligned
    const int edgeWaveBl = (ET + 7) / 8;            // wave-per-edge, 8 waves/blk

    // ---- layer 1 init ----
    fill_u32<<<(N * 4 + B - 1) / B, B, 0, stream>>>(sm1, MU_NEG_INF, N * 4);
    fill_u32<<<(N * 4 + B - 1) / B, B, 0, stream>>>((unsigned*)ss1, 0u, N * 4);
    fill_u32<<<(N * HC + B - 1) / B, B, 0, stream>>>((unsigned*)acc1, 0u, N * HC);

    // ---- layer 1: xl = x@w1_l, xr = x@w1_r (WMMA f32, K=128, ncol=128) ----
    gemm16_wmma_f32<F_IN, HC><<<rowTiles, B, 0, stream>>>(x, w1l, xl1);
    gemm16_wmma_f32<F_IN, HC><<<rowTiles, B, 0, stream>>>(x, w1r, xr1);

    // ---- layer 1: attention ----
    edge_logits_h4<<<edgeWaveBl, B, 0, stream>>>(ei, E, ET, xl1, xr1, a1, e1, sm1);
    expsum_h4<<<(ET * 4 + B - 1) / B, B, 0, stream>>>(ei, E, ET, e1, sm1, ss1);
    aggregate_h4<<<edgeWaveBl, B, 0, stream>>>(ei, E, ET, e1, ss1, xl1, acc1);
    elu_bias<<<(N * HC + B - 1) / B, B, 0, stream>>>(acc1, b1, N * HC);

    // ---- layer 2: hl = h@w2_l, hr = h@w2_r (WMMA f32, K=128, ncol=40) ----
    gemm16_wmma_f32<HC, CLS><<<rowTiles, B, 0, stream>>>(acc1, w2l, xl2);
    gemm16_wmma_f32<HC, CLS><<<rowTiles, B, 0, stream>>>(acc1, w2r, xr2);

    // ---- layer 2 init (layer-1 stat buffers are dead now) ----
    fill_u32<<<(N + B - 1) / B, B, 0, stream>>>(sm2, MU_NEG_INF, N);
    fill_u32<<<(N + B - 1) / B, B, 0, stream>>>((unsigned*)ss2, 0u, N);
    fill_u32<<<(N * CLS + B - 1) / B, B, 0, stream>>>((unsigned*)acc2, 0u, N * CLS);

    // ---- layer 2: attention ----
    edge_logits_h1<<<edgeWaveBl, B, 0, stream>>>(ei, E, ET, xl2, xr2, a2, e2, sm2);
    expsum_h1<<<(ET + B - 1) / B, B, 0, stream>>>(ei, E, ET, e2, sm2, ss2);
    aggregate_h1<<<edgeWaveBl, B, 0, stream>>>(ei, E, ET, e2, ss2, xl2, acc2);

    // ---- log_softmax -> d_out ----
    finalize_logsoftmax<<<(N + 7) / 8, B, 0, stream>>>(acc2, b2, out, N);
}